// DownSample_11390253269666
// MI455X (gfx1250) — compile-verified
//
#include <hip/hip_runtime.h>

typedef __attribute__((ext_vector_type(2))) float v2f;
typedef __attribute__((ext_vector_type(4))) float v4f;
typedef __attribute__((ext_vector_type(8))) float v8f;

#define COUT 64

// Gathered sparse conv:  OUT[r,:] = relu(bns * (sum_k X[idx(k,r),:] @ W[k]) + bnb)
// idx(k,r) = KMAJOR ? idxmap[k*n_out + r] : idxmap[r*NOFF + k];  idx >= n_src means "zero row".
// Block = 128 threads (4 waves). Each wave computes a 16-row x 64-col output tile with
// V_WMMA_F32_16X16X4_F32 (full fp32, matches reference precision).
template <int K, int NOFF, bool KMAJOR>
__global__ __launch_bounds__(128) void wmma_sparse_conv(
    const float* __restrict__ X,       // [n_src, K]
    const float* __restrict__ W,       // [NOFF, K, 64]
    const float* __restrict__ bns,     // [64]
    const float* __restrict__ bnb,     // [64]
    const int*   __restrict__ idxmap,
    float* __restrict__ OUT,           // [n_out, 64]
    int n_src, int n_out)
{
    constexpr int AS   = K + 4;        // sA row stride: banks {4*hl+c} vs {4*hl+c+2} disjoint
    constexpr int NFRG = (K / 4) * 4;  // B fragments per offset (kc-chunks x 4 N-tiles)

    // B fragments pre-swizzled: slot (frag, lane) = (B[kc+2*hi][nt*16+hl], B[kc+1+2*hi][...])
    __shared__ __align__(16) v2f   sB[NFRG * 32];
    __shared__ __align__(16) float sA[4][16 * AS];

    const int tid  = threadIdx.x;
    const int wave = tid >> 5;
    const int lane = tid & 31;
    const int hl   = lane & 15;        // lane within half-wave
    const int hi   = lane >> 4;        // which half-wave
    const int tile = blockIdx.x * 64;  // first output row of this block
    const int row  = tile + wave * 16 + hl;  // gather row owned by this lane pair

    v8f acc[4] = {};                   // 16x64 fp32 accumulator tile (4 N-tiles of 16)

    for (int k = 0; k < NOFF; ++k) {
        // ---- stage W[k] (K x 64) into fragment-ready LDS layout, b128 global loads ----
        for (int e = tid; e < K * COUT / 4; e += 128) {
            const int ci = e >> 4;          // input-channel row 0..K-1
            const int c0 = (e & 15) * 4;    // 4 consecutive output cols
            v4f w4 = *(const v4f*)(W + ((size_t)k * K + ci) * COUT + c0);
            const int kc4 = ci >> 2;        // which K-chunk
            const int j   = ci & 1;         // element within the lane's v2f pair
            const int hid = (ci >> 1) & 1;  // destination half-wave
            const int nt  = c0 >> 4;        // destination N-tile
            float* dst = (float*)&sB[(kc4 * 4 + nt) * 32 + hid * 16 + (c0 & 15)];
            dst[0 * 2 + j] = w4.x;
            dst[1 * 2 + j] = w4.y;
            dst[2 * 2 + j] = w4.z;
            dst[3 * 2 + j] = w4.w;
        }
        // ---- gather 16 activation rows per wave into LDS (zero-fill invalid) ----
        int src = n_src;
        if (row < n_out)
            src = KMAJOR ? idxmap[(size_t)k * n_out + row] : idxmap[(size_t)row * NOFF + k];
        const int cbase = hi * (K / 2);
        float* arow = &sA[wave][hl * AS + cbase];
        if ((unsigned)src < (unsigned)n_src) {
            const float* xr = X + (size_t)src * K + cbase;
#pragma unroll
            for (int c = 0; c < K / 2; c += 4) {
                v4f v = *(const v4f*)(xr + c);
                arow[c + 0] = v.x; arow[c + 1] = v.y;
                arow[c + 2] = v.z; arow[c + 3] = v.w;
            }
        } else {
#pragma unroll
            for (int c = 0; c < K / 2; ++c) arow[c] = 0.f;
        }
        __syncthreads();

        // ---- K/4 x 4 WMMAs: single aligned v2f DS load per fragment, no repacking ----
#pragma unroll
        for (int kc4 = 0; kc4 < K / 4; ++kc4) {
            v2f a = *(const v2f*)&sA[wave][hl * AS + kc4 * 4 + 2 * hi];
#pragma unroll
            for (int nt = 0; nt < 4; ++nt) {
                v2f b = sB[(kc4 * 4 + nt) * 32 + lane];
                acc[nt] = __builtin_amdgcn_wmma_f32_16x16x4_f32(
                    false, a, false, b, (short)0, acc[nt], false, false);
            }
        }
        __syncthreads();
    }

    // ---- fused BN + ReLU epilogue; C/D layout: vgpr v, half hi -> row v + 8*hi ----
    const int orow0 = tile + wave * 16;
#pragma unroll
    for (int nt = 0; nt < 4; ++nt) {
        const int col = nt * 16 + hl;
        const float s = bns[col];
        const float b = bnb[col];
#pragma unroll
        for (int v = 0; v < 8; ++v) {
            int r = orow0 + v + 8 * hi;
            if (r < n_out) {
                float val = acc[nt][v] * s + b;
                OUT[(size_t)r * COUT + col] = val > 0.f ? val : 0.f;
            }
        }
    }
}

// ---- invert kp_in/kp_out into dense parent -> child[8] gather map ----
__global__ void init_cmap(int* cmap, int n_entries, int sentinel) {
    int t = blockIdx.x * blockDim.x + threadIdx.x;
    if (t < n_entries) cmap[t] = sentinel;
}

__global__ void build_cmap(const int* __restrict__ kp_in, const int* __restrict__ kp_out,
                           int* __restrict__ cmap, int n, int n_down) {
    int t = blockIdx.x * blockDim.x + threadIdx.x;   // over 8*n
    if (t >= 8 * n) return;
    int k     = t / n;
    int child = kp_in[t];
    int par   = kp_out[t];
    if (child < n && par < n_down)   // each (parent,k) written at most once -> race-free
        cmap[par * 8 + k] = child;
}

extern "C" void kernel_launch(void* const* d_in, const int* in_sizes, int n_in,
                              void* d_out, int out_size, void* d_ws, size_t ws_size,
                              hipStream_t stream) {
    const float* x    = (const float*)d_in[0];
    const float* W1   = (const float*)d_in[1];
    const float* bn1s = (const float*)d_in[2];
    const float* bn1b = (const float*)d_in[3];
    const float* W2   = (const float*)d_in[4];
    const float* bn2s = (const float*)d_in[5];
    const float* bn2b = (const float*)d_in[6];
    const float* Wp   = (const float*)d_in[7];
    const float* bnps = (const float*)d_in[8];
    const float* bnpb = (const float*)d_in[9];
    const int* km_in  = (const int*)d_in[10];
    // d_in[11] = km_out: redundant (== i where km_in valid, N otherwise) in gather form
    const int* kp_in  = (const int*)d_in[12];
    const int* kp_out = (const int*)d_in[13];

    const int n      = in_sizes[0] / 32;  // C_IN = 32
    const int n_down = n / 4;             // CHILD = 4

    float* down = (float*)d_out;                       // [n, 64]
    float* p    = (float*)d_out + (size_t)n * COUT;    // [n_down, 64]
    float* h    = (float*)d_ws;                        // [n, 64] scratch
    int*   cmap = (int*)((char*)d_ws + (size_t)n * COUT * sizeof(float)); // [n_down*8]

    dim3 blk(128);
    const int cgrid = (n + 63) / 64;
    const int pgrid = (n_down + 63) / 64;

    // conv1: x[N,32] -> h[N,64]
    wmma_sparse_conv<32, 27, true><<<cgrid, blk, 0, stream>>>(x, W1, bn1s, bn1b, km_in, h, n, n);
    // conv2: h[N,64] -> down[N,64]  (written straight into d_out)
    wmma_sparse_conv<64, 27, true><<<cgrid, blk, 0, stream>>>(h, W2, bn2s, bn2b, km_in, down, n, n);
    // build parent->children map, then pool conv: down -> p[N_DOWN,64]
    const int ce = n_down * 8;
    init_cmap <<<(ce + 255) / 256, 256, 0, stream>>>(cmap, ce, n);
    build_cmap<<<(8 * n + 255) / 256, 256, 0, stream>>>(kp_in, kp_out, cmap, n, n_down);
    wmma_sparse_conv<64, 8, false><<<pgrid, blk, 0, stream>>>(down, Wp, bnps, bnpb, cmap, p, n, n_down);
}